// PropagationEncoder_89309549953741
// MI455X (gfx1250) — compile-verified
//
#include <hip/hip_runtime.h>
#include <math.h>

// ---------------------------------------------------------------------------
// CDNA5 (gfx1250) wave32 WMMA types
// ---------------------------------------------------------------------------
typedef __attribute__((ext_vector_type(2))) float v2f;  // A/B operand of 16x16x4 f32 WMMA
typedef __attribute__((ext_vector_type(8))) float v8f;  // 16x16 f32 accumulator

#define WMMA_F32X4(a, b, c) \
  __builtin_amdgcn_wmma_f32_16x16x4_f32(false, (a), false, (b), (short)0, (c), false, false)

#define KDIM 128  // all A@W^T GEMMs in this model have K == 128 per term

// ---------------------------------------------------------------------------
// Small utility kernels
// ---------------------------------------------------------------------------
__global__ void zero_f32(float* __restrict__ p, long long n) {
  long long i = (long long)blockIdx.x * blockDim.x + threadIdx.x;
  if (i < n) p[i] = 0.0f;
}

// root[g] = first index i with batch[i] == g (batch sorted, all graphs nonempty)
__global__ void find_roots(const int* __restrict__ batch, int N, int G,
                           int* __restrict__ root) {
  int g = blockIdx.x * blockDim.x + threadIdx.x;
  if (g >= G) return;
  int lo = 0, hi = N;
  while (lo < hi) {
    int mid = (lo + hi) >> 1;
    if (batch[mid] < g) lo = mid + 1; else hi = mid;
  }
  root[g] = lo;
}

// One wave32 per edge; each lane moves 4 of the 128 features as float4,
// accumulated into agg[dst] with native global_atomic_add_f32 (L2-resident).
__global__ void edge_agg(const int* __restrict__ ei, int E,
                         const float* __restrict__ x,
                         float* __restrict__ agg, float* __restrict__ deg) {
  long long t = (long long)blockIdx.x * blockDim.x + threadIdx.x;
  long long e = t >> 5;
  int lane = (int)(t & 31);
  if (e >= (long long)E) return;
  int src = ei[e];
  int dst = ei[(long long)E + e];
  const float4 v = ((const float4*)(x + (size_t)src * 128))[lane];
  float* ap = agg + (size_t)dst * 128 + lane * 4;
  unsafeAtomicAdd(ap + 0, v.x);
  unsafeAtomicAdd(ap + 1, v.y);
  unsafeAtomicAdd(ap + 2, v.z);
  unsafeAtomicAdd(ap + 3, v.w);
  if (lane == 0) unsafeAtomicAdd(deg + dst, 1.0f);
}

__global__ void inv_deg(const float* __restrict__ deg, float* __restrict__ invd, int N) {
  int i = blockIdx.x * blockDim.x + threadIdx.x;
  if (i < N) invd[i] = 1.0f / fmaxf(deg[i], 1.0f);
}

// Per-graph feature-wise max over sorted node ranges. Grid = G, block = H.
__global__ void seg_max(const float* __restrict__ p, const int* __restrict__ root,
                        int N, int G, int H, float* __restrict__ pmax) {
  int g = blockIdx.x;
  int f = threadIdx.x;
  int s = root[g];
  int e = (g == G - 1) ? N : root[g + 1];
  float m = -INFINITY;
  for (int r = s; r < e; ++r) m = fmaxf(m, p[(size_t)r * H + f]);
  pmax[(size_t)g * H + f] = m;
}

// out[0:G*H] = v[slices[i]] rows ; out[G*H:] = e flat
__global__ void write_out(const float* __restrict__ v, const int* __restrict__ slices,
                          const float* __restrict__ e, int G, int M, int H,
                          float* __restrict__ out) {
  int i = blockIdx.x * blockDim.x + threadIdx.x;
  int gh = G * H;
  int total = gh + M * H;
  if (i >= total) return;
  if (i < gh) {
    int r = i / H, f = i - r * H;
    out[i] = v[(size_t)slices[r] * H + f];
  } else {
    out[i] = e[i - gh];
  }
}

// ---------------------------------------------------------------------------
// Fill LDS with a 128x128 K-slice of W, pre-permuted into the WMMA B-operand
// order: lwB[(j*32 + step)*32 + lane] = { W[j*16+nl][wk+4*step+koff],
//                                         W[j*16+nl][wk+4*step+koff+1] }
// where nl = lane&15, koff = 2*(lane>>4).  Reads in the compute loop are then
// 32 consecutive float2 per (j,step): one ds_load_b64 base register, 16-bit
// immediate offsets (max 7*8192+31*256 = 65280), all 64 LDS banks hit.
// ---------------------------------------------------------------------------
__device__ __forceinline__ void fill_w_lds(v2f* __restrict__ lwB,
                                           const float* __restrict__ W,
                                           int ldw, int wk, int tid) {
#pragma unroll
  for (int it = 0; it < 32; ++it) {
    int i = tid + it * 256;          // 0..8191, blockDim.x == 256
    int lane = i & 31;
    int step = (i >> 5) & 31;
    int j = i >> 10;
    int nl = lane & 15;
    int koff = (lane >> 4) << 1;
    int row = j * 16 + nl;
    int col = (step << 2) + koff;
    lwB[i] = *(const v2f*)(W + (size_t)row * ldw + wk + col);
  }
}

// ---------------------------------------------------------------------------
// WMMA GEMM #1: out[m, 0:128] = relu( s1[g1(m)]*A1[g1(m),:] @ W1[:, wk1:+128]^T
//                                    + A2[m,:] @ W2[:, wk2:+128]^T + bias )
// One wave computes a 16x128 output tile (8 v8f accumulators); W staged in
// 64KB LDS in B-operand order; K fully unrolled (immediate-offset loads only).
// No early wave exit: tail waves run with clamped loads / guarded stores so
// block barriers stay valid and EXEC stays all-ones for WMMA.
// ---------------------------------------------------------------------------
__global__ __launch_bounds__(256) void gemm_awt_bias_relu(
    const float* __restrict__ A1, int lda1,
    const int* __restrict__ idx1, const float* __restrict__ rowscale1,
    const float* __restrict__ W1, int ldw1, int wk1,
    const float* __restrict__ A2, int lda2,
    const float* __restrict__ W2, int ldw2, int wk2,
    const float* __restrict__ bias,
    float* __restrict__ out, int ldo, int Mrows) {
  __shared__ v2f lwB[8192];  // 64 KB

  const int tid = threadIdx.x;
  const int wave = tid >> 5;
  const int lane = tid & 31;
  const int row0 = (blockIdx.x * 8 + wave) * 16;

  const int nl = lane & 15;
  const int koff = (lane >> 4) << 1;  // 0 or 2
  const int mrow = row0 + nl;
  const int mc = mrow < Mrows ? mrow : (Mrows - 1);  // clamp tail-row loads
  const int r1 = idx1 ? idx1[mc] : mc;
  const float sc = rowscale1 ? rowscale1[r1] : 1.0f;

  v8f acc[8] = {};

  // ---- term 1 ----
  fill_w_lds(lwB, W1, ldw1, wk1, tid);
  __syncthreads();
  {
    const float* a1p = A1 + (size_t)r1 * lda1 + koff;
#pragma unroll
    for (int step = 0; step < KDIM / 4; ++step) {
      v2f a = *(const v2f*)(a1p + step * 4);
      a *= sc;
#pragma unroll
      for (int j = 0; j < 8; ++j) {
        v2f b = lwB[(j * 32 + step) * 32 + lane];
        acc[j] = WMMA_F32X4(a, b, acc[j]);
      }
    }
  }

  // ---- term 2 (optional) ----
  if (A2) {
    __syncthreads();
    fill_w_lds(lwB, W2, ldw2, wk2, tid);
    __syncthreads();
    const float* a2p = A2 + (size_t)mc * lda2 + koff;
#pragma unroll
    for (int step = 0; step < KDIM / 4; ++step) {
      v2f a = *(const v2f*)(a2p + step * 4);
#pragma unroll
      for (int j = 0; j < 8; ++j) {
        v2f b = lwB[(j * 32 + step) * 32 + lane];
        acc[j] = WMMA_F32X4(a, b, acc[j]);
      }
    }
  }

  // ---- epilogue: D vgpr r -> row0 + r + 8*(lane>=16), col = 16*j + nl ----
  float bval[8];
#pragma unroll
  for (int j = 0; j < 8; ++j) bval[j] = bias[j * 16 + nl];

  const int rbase = row0 + ((lane >> 4) << 3);
#pragma unroll
  for (int r = 0; r < 8; ++r) {
    int row = rbase + r;
    if (row < Mrows) {
#pragma unroll
      for (int j = 0; j < 8; ++j) {
        float val = acc[j][r] + bval[j];
        out[(size_t)row * ldo + j * 16 + nl] = fmaxf(val, 0.0f);
      }
    }
  }
}

// ---------------------------------------------------------------------------
// WMMA GEMM #2:  out = A @ B  (B row-major [K,128]; A row-major or A^T view)
// transA: A[m][k] = A[k*lda + m]   (used for HT^T @ e). Tiny GEMMs only.
// ---------------------------------------------------------------------------
__global__ __launch_bounds__(256) void gemm_ab(
    const float* __restrict__ A, int lda, int transA,
    const float* __restrict__ B, int ldb,
    float* __restrict__ out, int ldo, int Mrows, int K) {
  const int wave = threadIdx.x >> 5;
  const int lane = threadIdx.x & 31;
  const int row0 = (blockIdx.x * 8 + wave) * 16;
  if (row0 >= Mrows) return;  // wave-uniform (no block barriers here)

  const int nl = lane & 15;
  const int koff = (lane >> 4) << 1;
  const int mrow = row0 + nl;
  const int mc = mrow < Mrows ? mrow : (Mrows - 1);

  v8f acc[8] = {};

  for (int k = 0; k < K; k += 4) {
    v2f a;
    if (transA) {
      a.x = A[(size_t)(k + koff) * lda + mc];
      a.y = A[(size_t)(k + koff + 1) * lda + mc];
    } else {
      a = *(const v2f*)(A + (size_t)mc * lda + k + koff);
    }
#pragma unroll
    for (int j = 0; j < 8; ++j) {
      v2f b;
      b.x = B[(size_t)(k + koff) * ldb + j * 16 + nl];
      b.y = B[(size_t)(k + koff + 1) * ldb + j * 16 + nl];
      acc[j] = WMMA_F32X4(a, b, acc[j]);
    }
  }

  const int rbase = row0 + ((lane >> 4) << 3);
#pragma unroll
  for (int r = 0; r < 8; ++r) {
    int row = rbase + r;
    if (row < Mrows) {
#pragma unroll
      for (int j = 0; j < 8; ++j) {
        out[(size_t)row * ldo + j * 16 + nl] = acc[j][r];
      }
    }
  }
}

// ---------------------------------------------------------------------------
// Host-side orchestration
// ---------------------------------------------------------------------------
static inline long long cdiv(long long a, long long b) { return (a + b - 1) / b; }

extern "C" void kernel_launch(void* const* d_in, const int* in_sizes, int n_in,
                              void* d_out, int out_size, void* d_ws, size_t ws_size,
                              hipStream_t stream) {
  (void)n_in; (void)out_size; (void)ws_size;

  const float* x        = (const float*)d_in[0];
  const float* HT       = (const float*)d_in[1];
  const float* lin0_w   = (const float*)d_in[2];
  const float* lin0_b   = (const float*)d_in[3];
  const float* sage_l_w = (const float*)d_in[4];
  const float* sage_l_b = (const float*)d_in[5];
  const float* sage_r_w = (const float*)d_in[6];
  const float* lin1_w   = (const float*)d_in[7];
  const float* lin1_b   = (const float*)d_in[8];
  const float* hg_We    = (const float*)d_in[9];
  const float* hg_be    = (const float*)d_in[10];
  const float* hg_Wv    = (const float*)d_in[11];
  const float* hg_bv    = (const float*)d_in[12];
  const int*   edge_idx = (const int*)d_in[13];
  const int*   batch    = (const int*)d_in[14];
  const int*   slices   = (const int*)d_in[15];

  const int N = in_sizes[14];            // 100000
  const int F = in_sizes[0] / N;         // 128
  const int H = in_sizes[3];             // 128
  const int E = in_sizes[13] / 2;        // 1.6M
  const int G = in_sizes[15];            // 1000
  const int M = in_sizes[1] / G;         // 256

  // -------- workspace carving (256B-aligned slabs) --------
  char* ws = (char*)d_ws;
  size_t off = 0;
  auto takeF = [&](size_t nelem) -> float* {
    float* p = (float*)(ws + off);
    off += ((nelem * sizeof(float)) + 255) & ~(size_t)255;
    return p;
  };
  float* agg  = takeF((size_t)N * F);
  float* p    = takeF((size_t)N * H);
  float* deg  = takeF((size_t)N);
  float* invd = takeF((size_t)N);
  float* news = takeF((size_t)G * H);
  float* pmax = takeF((size_t)G * H);
  float* p2   = takeF((size_t)G * H);
  float* tbuf = takeF((size_t)M * H);
  float* ebuf = takeF((size_t)M * H);
  float* vpre = takeF((size_t)G * H);
  float* vbuf = takeF((size_t)G * H);
  int*   root = (int*)takeF((size_t)G);

  // -------- 1. init accumulators (must be re-zeroed every call) --------
  {
    long long n = (long long)N * F;
    zero_f32<<<(unsigned)cdiv(n, 256), 256, 0, stream>>>(agg, n);
    zero_f32<<<(unsigned)cdiv(N, 256), 256, 0, stream>>>(deg, N);
  }

  // -------- 2. roots of each graph --------
  find_roots<<<(unsigned)cdiv(G, 256), 256, 0, stream>>>(batch, N, G, root);

  // -------- 3. edge aggregation (dominant, L2-resident atomics) --------
  {
    long long threads = (long long)E * 32;
    edge_agg<<<(unsigned)cdiv(threads, 256), 256, 0, stream>>>(edge_idx, E, x, agg, deg);
  }
  inv_deg<<<(unsigned)cdiv(N, 256), 256, 0, stream>>>(deg, invd, N);

  // -------- 4. p = relu((agg*invd) @ Wl^T + x @ Wr^T + b)  [N,128] --------
  gemm_awt_bias_relu<<<(unsigned)cdiv(N, 128), 256, 0, stream>>>(
      agg, F, nullptr, invd, sage_l_w, F, 0,
      x, F, sage_r_w, F, 0, sage_l_b, p, H, N);

  // -------- 5. news = relu(x[root] @ lin0_w^T + b)  [G,128] --------
  gemm_awt_bias_relu<<<(unsigned)cdiv(G, 128), 256, 0, stream>>>(
      x, F, root, nullptr, lin0_w, F, 0,
      nullptr, 0, nullptr, 0, 0, lin0_b, news, H, G);

  // -------- 6. pmax = segment_max(p, batch)  [G,128] --------
  seg_max<<<G, H, 0, stream>>>(p, root, N, G, H, pmax);

  // -------- 7. p2 = relu([news|pmax] @ lin1_w^T + b)  [G,128] --------
  gemm_awt_bias_relu<<<(unsigned)cdiv(G, 128), 256, 0, stream>>>(
      news, H, nullptr, nullptr, lin1_w, 2 * H, 0,
      pmax, H, lin1_w, 2 * H, H, lin1_b, p2, H, G);

  // -------- 8. t = HT @ p2  [M,128], K=G (divisible by 4) --------
  gemm_ab<<<(unsigned)cdiv(M, 128), 256, 0, stream>>>(HT, G, 0, p2, H, tbuf, H, M, G);

  // -------- 9. e = relu(t @ We^T + be)  [M,128] --------
  gemm_awt_bias_relu<<<(unsigned)cdiv(M, 128), 256, 0, stream>>>(
      tbuf, H, nullptr, nullptr, hg_We, H, 0,
      nullptr, 0, nullptr, 0, 0, hg_be, ebuf, H, M);

  // -------- 10. vpre = HT^T @ e  [G,128], K=M --------
  gemm_ab<<<(unsigned)cdiv(G, 128), 256, 0, stream>>>(HT, G, 1, ebuf, H, vpre, H, G, M);

  // -------- 11. v = relu(vpre @ Wv^T + bv)  [G,128] --------
  gemm_awt_bias_relu<<<(unsigned)cdiv(G, 128), 256, 0, stream>>>(
      vpre, H, nullptr, nullptr, hg_Wv, H, 0,
      nullptr, 0, nullptr, 0, 0, hg_bv, vbuf, H, G);

  // -------- 12. out = [v[slices] ; e] --------
  {
    int total = G * H + M * H;
    write_out<<<(unsigned)cdiv(total, 256), 256, 0, stream>>>(
        vbuf, slices, ebuf, G, M, H, (float*)d_out);
  }
}